// FFT_19490561589800
// MI455X (gfx1250) — compile-verified
//
#include <hip/hip_runtime.h>

typedef float v2f __attribute__((ext_vector_type(2)));
typedef float v8f __attribute__((ext_vector_type(8)));

#define N_FFT   1024
#define ROWS    16
#define THREADS 256

// ---- LDS layout (units: floats) ----
// xs: permuted input, [16 rows][16 chunks][64+4 pad] (chunk-major; pad breaks
//     the 64-bank periodicity for the scattered async writes)
#define CHUNK_STR 68
#define XS_ROW    (16 * CHUNK_STR)          // 1088
#define XS_OFF    0
#define XS_SZ     (ROWS * XS_ROW)           // 17408
// F: DFT matrices ROW-MAJOR, stride 66 (even -> b64-aligned B fragments,
//    66 mod 64 == 2 -> at most 2-way bank conflict across the 16 N-lanes)
#define FT_STR    66
#define FR_OFF    (XS_OFF + XS_SZ)          // 17408
#define FI_OFF    (FR_OFF + 64 * FT_STR)    // 21632
// twiddles, stages packed at offsets 0,64,192,448 (len 64,128,256,512)
#define TWR_OFF   (FI_OFF + 64 * FT_STR)    // 25856
#define TWI_OFF   (TWR_OFF + 960)           // 26816
// complex state, [16 rows][1024], in-place butterfly workspace
#define STR_OFF   (TWI_OFF + 960)           // 27776
#define STI_OFF   (STR_OFF + ROWS * N_FFT)  // 44160
#define LDS_FLOATS (STI_OFF + ROWS * N_FFT) // 60544 -> 242,176 B < 320 KB

__device__ __forceinline__ int bitrev4(int o) {
    return ((o & 1) << 3) | ((o & 2) << 1) | ((o & 4) >> 1) | ((o & 8) >> 3);
}

__device__ __forceinline__ v8f wmma_f32_4(v2f a, v2f b, v8f c) {
    return __builtin_amdgcn_wmma_f32_16x16x4_f32(false, a, false, b,
                                                 (short)0, c, false, false);
}

__global__ void __launch_bounds__(THREADS)
fft1024_wmma_kernel(const float* __restrict__ x,
                    const float* __restrict__ fft_real,
                    const float* __restrict__ fft_imag,
                    const float* __restrict__ twr0, const float* __restrict__ twi0,
                    const float* __restrict__ twr1, const float* __restrict__ twi1,
                    const float* __restrict__ twr2, const float* __restrict__ twi2,
                    const float* __restrict__ twr3, const float* __restrict__ twi3,
                    float* __restrict__ out, int batch)
{
    extern __shared__ float lds[];
    const int t    = threadIdx.x;
    const int row0 = blockIdx.x * ROWS;

    // ---- Phase 1a: async gather-permute of 16 input rows into LDS ----
    // Permuted column c*64+j <- x[:, j*16 + bitrev4(c)]: coalesced global
    // reads, per-lane scattered LDS destinations -- exactly what
    // GLOBAL_LOAD_ASYNC_TO_LDS_B32 (per-lane LDS dest VGPR) supports.
    {
        const float* xbase = x + (size_t)row0 * N_FFT;           // uniform
        const unsigned lds_base = __builtin_amdgcn_groupstaticsize();
        const int o = t & 15;                 // source col mod 16: invariant
        const int c = bitrev4(o);             // destination chunk: invariant
        #pragma unroll 8
        for (int i = 0; i < (ROWS * N_FFT) / THREADS; ++i) {
            int e = t + i * THREADS;
            int r = e >> 10;
            int j = (e & 1023) >> 4;
            unsigned voff    = (unsigned)e * 4u;                 // global byte off
            unsigned ldsaddr = lds_base +
                4u * (unsigned)(XS_OFF + r * XS_ROW + c * CHUNK_STR + j);
            asm volatile("global_load_async_to_lds_b32 %0, %1, %2"
                         :: "v"(ldsaddr), "v"(voff), "s"(xbase) : "memory");
        }
    }

    // ---- Phase 1b: DFT matrices row-major (B[k][n] = F[n][k] is then a
    //      contiguous 8-byte pair per lane) ----
    #pragma unroll
    for (int i = 0; i < 4096 / THREADS; ++i) {
        int e = t + i * THREADS;
        int row = e >> 6, s = e & 63;
        lds[FR_OFF + row * FT_STR + s] = fft_real[e];
        lds[FI_OFF + row * FT_STR + s] = fft_imag[e];
    }
    // ---- Phase 1c: twiddles ----
    for (int k = t; k < 64;  k += THREADS) { lds[TWR_OFF +   0 + k] = twr0[k]; lds[TWI_OFF +   0 + k] = twi0[k]; }
    for (int k = t; k < 128; k += THREADS) { lds[TWR_OFF +  64 + k] = twr1[k]; lds[TWI_OFF +  64 + k] = twi1[k]; }
    for (int k = t; k < 256; k += THREADS) { lds[TWR_OFF + 192 + k] = twr2[k]; lds[TWI_OFF + 192 + k] = twi2[k]; }
    for (int k = t; k < 512; k += THREADS) { lds[TWR_OFF + 448 + k] = twr3[k]; lds[TWI_OFF + 448 + k] = twi3[k]; }

    asm volatile("s_wait_asynccnt 0x0" ::: "memory");   // drain async -> LDS
    __syncthreads();

    // ---- Phase 2: 16x 64-point DFT via V_WMMA_F32_16X16X4_F32 ----
    // Wave w owns chunks 2w, 2w+1. A fragments for both chunks are cached in
    // registers (2 x 16 x v2f); each B b64 load then feeds two WMMAs.
    {
        const int lane = t & 31;
        const int wave = t >> 5;
        const int M    = lane & 15;           // A row (lanes 0-15 / 16-31)
        const int kb   = (lane >> 4) << 1;    // K base: 0 or 2
        const int nl   = lane & 15;           // B/D column
        const int c0   = wave * 2, c1 = c0 + 1;

        v2f a0[16], a1[16];
        const float* A0 = &lds[XS_OFF + M * XS_ROW + c0 * CHUNK_STR + kb];
        const float* A1 = &lds[XS_OFF + M * XS_ROW + c1 * CHUNK_STR + kb];
        #pragma unroll
        for (int kk = 0; kk < 16; ++kk) {
            a0[kk] = *(const v2f*)&A0[kk * 4];
            a1[kk] = *(const v2f*)&A1[kk * 4];
        }

        const int rbase = (lane >> 4) << 3;   // D rows: v or v+8
        for (int ct = 0; ct < 8; ++ct) {      // comp*4 + ntile
            const int comp  = ct >> 2;
            const int ntile = ct & 3;
            const float* Fb = &lds[(comp ? FI_OFF : FR_OFF) +
                                   (ntile * 16 + nl) * FT_STR + kb];
            v8f acc0 = {0.f,0.f,0.f,0.f,0.f,0.f,0.f,0.f};
            v8f acc1 = {0.f,0.f,0.f,0.f,0.f,0.f,0.f,0.f};
            #pragma unroll
            for (int kk = 0; kk < 16; ++kk) {
                v2f b = *(const v2f*)&Fb[kk * 4];   // one ds_load_b64
                acc0 = wmma_f32_4(a0[kk], b, acc0);
                acc1 = wmma_f32_4(a1[kk], b, acc1);
            }
            float* st = &lds[comp ? STI_OFF : STR_OFF];
            const int col0 = c0 * 64 + ntile * 16 + nl;
            const int col1 = c1 * 64 + ntile * 16 + nl;
            #pragma unroll
            for (int v = 0; v < 8; ++v) {
                st[(rbase + v) * N_FFT + col0] = acc0[v];
                st[(rbase + v) * N_FFT + col1] = acc1[v];
            }
        }
    }
    __syncthreads();

    // ---- Phase 3: four in-place radix-2 stages (h = 64,128,256,512),
    //      vectorized: two adjacent butterflies (same block, k even) ----
    #pragma unroll
    for (int s = 0; s < 4; ++s) {
        const int lh = 6 + s;
        const int h  = 1 << lh;
        const int twoff = h - 64;
        #pragma unroll
        for (int i = 0; i < (ROWS * 256) / THREADS; ++i) {
            int idx = t + i * THREADS;          // 4096 float2 butterflies
            int r  = idx >> 8;
            int q2 = (idx & 255) << 1;          // even butterfly index
            int k  = q2 & (h - 1);
            int pos = ((q2 >> lh) << (lh + 1)) + k;
            float* sr = &lds[STR_OFF + r * N_FFT];
            float* si = &lds[STI_OFF + r * N_FFT];
            float2 tw_r = *(const float2*)&lds[TWR_OFF + twoff + k];
            float2 tw_i = *(const float2*)&lds[TWI_OFF + twoff + k];
            float2 er = *(float2*)&sr[pos],     ei = *(float2*)&si[pos];
            float2 odr = *(float2*)&sr[pos + h], odi = *(float2*)&si[pos + h];
            float2 br, bi;
            br.x = tw_r.x * odr.x - tw_i.x * odi.x;
            br.y = tw_r.y * odr.y - tw_i.y * odi.y;
            bi.x = tw_r.x * odi.x + tw_i.x * odr.x;
            bi.y = tw_r.y * odi.y + tw_i.y * odr.y;
            float2 u;
            u.x = er.x + br.x; u.y = er.y + br.y; *(float2*)&sr[pos] = u;
            u.x = ei.x + bi.x; u.y = ei.y + bi.y; *(float2*)&si[pos] = u;
            u.x = er.x - br.x; u.y = er.y - br.y; *(float2*)&sr[pos + h] = u;
            u.x = ei.x - bi.x; u.y = ei.y - bi.y; *(float2*)&si[pos + h] = u;
        }
        __syncthreads();
    }

    // ---- Phase 4: float4 stores, real plane then imag plane ----
    const size_t imag_base = (size_t)batch * N_FFT;
    #pragma unroll
    for (int i = 0; i < (ROWS * N_FFT / 4) / THREADS; ++i) {
        int e = t + i * THREADS;
        int r = e >> 8;
        int col = (e & 255) << 2;
        float4 vr = *(const float4*)&lds[STR_OFF + r * N_FFT + col];
        float4 vi = *(const float4*)&lds[STI_OFF + r * N_FFT + col];
        size_t gi = (size_t)(row0 + r) * N_FFT + col;
        *(float4*)&out[gi] = vr;
        *(float4*)&out[imag_base + gi] = vi;
    }
}

extern "C" void kernel_launch(void* const* d_in, const int* in_sizes, int n_in,
                              void* d_out, int out_size, void* d_ws, size_t ws_size,
                              hipStream_t stream) {
    const float* x  = (const float*)d_in[0];
    // d_in[1] = perm: applied analytically, matrix unused.
    const float* fr = (const float*)d_in[2];
    const float* fi = (const float*)d_in[3];
    const float* twr0 = (const float*)d_in[4];
    const float* twi0 = (const float*)d_in[5];
    const float* twr1 = (const float*)d_in[6];
    const float* twi1 = (const float*)d_in[7];
    const float* twr2 = (const float*)d_in[8];
    const float* twi2 = (const float*)d_in[9];
    const float* twr3 = (const float*)d_in[10];
    const float* twi3 = (const float*)d_in[11];
    float* out = (float*)d_out;

    const int batch = in_sizes[0] / N_FFT;
    const size_t ldsBytes = (size_t)LDS_FLOATS * sizeof(float);

    (void)hipFuncSetAttribute((const void*)fft1024_wmma_kernel,
                              hipFuncAttributeMaxDynamicSharedMemorySize,
                              (int)ldsBytes);

    dim3 grid(batch / ROWS), block(THREADS);
    fft1024_wmma_kernel<<<grid, block, ldsBytes, stream>>>(
        x, fr, fi, twr0, twi0, twr1, twi1, twr2, twi2, twr3, twi3, out, batch);
}